// RNMF_Encoder_simplified_85478439125016
// MI455X (gfx1250) — compile-verified
//
#include <hip/hip_runtime.h>

typedef __attribute__((ext_vector_type(2))) float v2f;
typedef __attribute__((ext_vector_type(4))) float v4f;
typedef __attribute__((ext_vector_type(8))) float v8f;

#define M_DIM   256
#define N_DIM   65536
#define Q_DIM   64
#define MQ      320            // m + q
#define LAM     0.1f
#define NT      64             // columns per workgroup
#define T_ITERS 16
#define PM_ITERS 100

// LDS holds U in WMMA-B operand order: element (k, col) lives at float index
//   ((k >> 2) * NT + col) * 4 + (k & 3)
// -> B operand (rows kr, kr+1 at fixed col) is one aligned b64 load; a 4-row
//    group at one col is one aligned b128 store. 80 k-groups x 64 cols.
#define UIDX(k4, col) (((k4) * NT + (col)) * 4)

// workspace layout in floats: [0]=eta, [1]=alpha, then H (320x320), then W (320x256)
#define H_OFF 16
#define W_OFF (H_OFF + MQ * MQ)

// ---------------------------------------------------------------------------
// Kernel 1: power method on D^T D  ->  eta = 1/alpha.  Tiny, one block.
// ---------------------------------------------------------------------------
__global__ __launch_bounds__(256) void pm_kernel(const float* __restrict__ D,
                                                 float* __restrict__ ws) {
  __shared__ float xv[Q_DIM];
  __shared__ float uv[M_DIM];
  __shared__ float vv[Q_DIM];
  __shared__ float nrm;
  const int tid = threadIdx.x;
  if (tid < Q_DIM) xv[tid] = 1.0f;
  __syncthreads();
  for (int it = 0; it < PM_ITERS; ++it) {
    float s = 0.f;
    const float* drow = D + tid * Q_DIM;
    for (int q = 0; q < Q_DIM; ++q) s += drow[q] * xv[q];
    uv[tid] = s;
    __syncthreads();
    if (tid < Q_DIM) {
      float t = 0.f;
      for (int i = 0; i < M_DIM; ++i) t += D[i * Q_DIM + tid] * uv[i];
      vv[tid] = t;
    }
    __syncthreads();
    if (tid == 0) {
      float ss = 0.f;
      for (int q = 0; q < Q_DIM; ++q) ss += vv[q] * vv[q];
      nrm = sqrtf(ss);
    }
    __syncthreads();
    if (tid < Q_DIM) xv[tid] = vv[tid] / nrm;
    __syncthreads();
  }
  if (tid == 0) { ws[1] = nrm; ws[0] = 1.0f / nrm; }
}

// ---------------------------------------------------------------------------
// Kernel 2: build H (320x320) and W (320x256), both with eta folded in.
// ---------------------------------------------------------------------------
__global__ __launch_bounds__(256) void build_hw_kernel(const float* __restrict__ D,
                                                       float* __restrict__ ws) {
  const int idx = blockIdx.x * blockDim.x + threadIdx.x;
  const float eta = ws[0];
  if (idx < MQ * MQ) {
    const int r = idx / MQ, c = idx % MQ;
    const float dlt = (r == c) ? 1.f : 0.f;
    float a;
    if (r < Q_DIM && c < Q_DIM) {
      float g = 0.f;
      for (int i = 0; i < M_DIM; ++i) g += D[i * Q_DIM + r] * D[i * Q_DIM + c];
      a = g + LAM * dlt;                        // D^T D + lam* I
    } else if (r < Q_DIM) {
      a = D[(c - Q_DIM) * Q_DIM + r];           // D^T block
    } else if (c < Q_DIM) {
      a = D[(r - Q_DIM) * Q_DIM + c];           // D block
    } else {
      a = (1.f + LAM) * dlt;                    // (1+lam*) I
    }
    ws[H_OFF + idx] = dlt - eta * a;            // H = I - eta * [...]
  } else if (idx < MQ * MQ + MQ * M_DIM) {
    const int j = idx - MQ * MQ;
    const int r = j / M_DIM, c = j % M_DIM;
    float wv;
    if (r < Q_DIM) wv = D[c * Q_DIM + r];       // D^T
    else           wv = (r - Q_DIM == c) ? 1.f : 0.f;  // I_m
    ws[W_OFF + j] = eta * wv;                   // W = eta * [D^T; I]
  }
}

// ---------------------------------------------------------------------------
// Kernel 3: fused b0 = W@X  +  16 ISTA iterations, all state resident.
// 20 waves/block; wave w owns rows [16w,16w+16) x 64 cols in WMMA C/D layout.
// ---------------------------------------------------------------------------
__global__ __launch_bounds__(640) void ista_kernel(const float* __restrict__ X,
                                                   const float* __restrict__ ws,
                                                   float* __restrict__ Z) {
  extern __shared__ float U[];                  // 80 * 64 * 4 floats (81920 B)
  const float* __restrict__ H = ws + H_OFF;
  const float* __restrict__ W = ws + W_OFF;
  const int tid  = threadIdx.x;
  const int w    = tid >> 5;                    // wave id (wave32)
  const int lane = tid & 31;
  const int li   = lane & 15;
  const int hi   = lane >> 4;
  const int cb0  = blockIdx.x * NT;
  const int rb   = w * 16;
  const float eta = ws[0];
  const float tw  = (rb >= Q_DIM) ? (LAM * eta) : 0.f;  // threshold, uniform per wave

  // Stage X tile (256 x 64) into LDS k-groups [0,64): one b128 store per group/col
  for (int idx = tid; idx < (M_DIM / 4) * NT; idx += 640) {
    const int g = idx >> 6, c = idx & (NT - 1);
    const float* xp = X + (4 * g) * N_DIM + cb0 + c;
    v4f t;
    t.x = xp[0];
    t.y = xp[N_DIM];
    t.z = xp[2 * N_DIM];
    t.w = xp[3 * N_DIM];
    *(v4f*)&U[UIDX(g, c)] = t;
  }
  __syncthreads();

  const v8f zero = {};
  v8f b[4], z[4];
#pragma unroll
  for (int cb = 0; cb < 4; ++cb) { b[cb] = zero; z[cb] = zero; }

  // -------- b0 = W @ Xtile : K = 256 -> 16 k-tiles of 16 --------
  const float* Wrow = W + (rb + li) * M_DIM + 2 * hi;   // A-operand base (16x4 layout)
  for (int kt = 0; kt < 16; ++kt) {
    v2f a[4];
#pragma unroll
    for (int kk = 0; kk < 4; ++kk)
      a[kk] = *(const v2f*)(Wrow + kt * 16 + kk * 4);   // 8B-aligned float2
#pragma unroll
    for (int kk = 0; kk < 4; ++kk) {
      const int k4 = kt * 4 + kk;
#pragma unroll
      for (int cb = 0; cb < 4; ++cb) {
        const v2f bb = *(const v2f*)&U[UIDX(k4, cb * 16 + li) + 2 * hi];
        b[cb] = __builtin_amdgcn_wmma_f32_16x16x4_f32(
            false, a[kk], false, bb, (short)0, b[cb], false, false);
      }
    }
  }
  __syncthreads();   // done reading X tile; U will be overwritten

  // -------- ISTA iterations --------
  const float* Hrow = H + (rb + li) * MQ + 2 * hi;
  for (int it = 0; it < T_ITERS; ++it) {
    // Znew = relu(b - t); U = Znew - Zold; Z = Znew
#pragma unroll
    for (int cb = 0; cb < 4; ++cb) {
      v8f zn;
#pragma unroll
      for (int v = 0; v < 8; ++v) zn[v] = fmaxf(0.f, b[cb][v] - tw);
      if (it < T_ITERS - 1) {
        const int col = cb * 16 + li;
        const int rg0 = (rb + 8 * hi) >> 2;     // first 4-row group this lane owns
#pragma unroll
        for (int g = 0; g < 2; ++g) {
          v4f u;
          u.x = zn[4 * g + 0] - z[cb][4 * g + 0];
          u.y = zn[4 * g + 1] - z[cb][4 * g + 1];
          u.z = zn[4 * g + 2] - z[cb][4 * g + 2];
          u.w = zn[4 * g + 3] - z[cb][4 * g + 3];
          *(v4f*)&U[UIDX(rg0 + g, col)] = u;
        }
      }
      z[cb] = zn;
    }
    if (it == T_ITERS - 1) break;               // final b update is dead
    __syncthreads();                            // U published

    // b += H @ U : K = 320 -> 20 k-tiles
    for (int kt = 0; kt < 20; ++kt) {
      if (kt < 19) __builtin_prefetch(Hrow + (kt + 1) * 16, 0, 1);
      v2f a[4];
#pragma unroll
      for (int kk = 0; kk < 4; ++kk)
        a[kk] = *(const v2f*)(Hrow + kt * 16 + kk * 4);
#pragma unroll
      for (int kk = 0; kk < 4; ++kk) {
        const int k4 = kt * 4 + kk;
#pragma unroll
        for (int cb = 0; cb < 4; ++cb) {
          const v2f bb = *(const v2f*)&U[UIDX(k4, cb * 16 + li) + 2 * hi];
          b[cb] = __builtin_amdgcn_wmma_f32_16x16x4_f32(
              false, a[kk], false, bb, (short)0, b[cb], false, false);
        }
      }
    }
    __syncthreads();                            // done reading U
  }

  // -------- write Z (320 x 65536, row-major) --------
#pragma unroll
  for (int cb = 0; cb < 4; ++cb)
#pragma unroll
    for (int v = 0; v < 8; ++v)
      Z[(rb + v + 8 * hi) * N_DIM + cb0 + cb * 16 + li] = z[cb][v];
}

// ---------------------------------------------------------------------------
extern "C" void kernel_launch(void* const* d_in, const int* in_sizes, int n_in,
                              void* d_out, int out_size, void* d_ws, size_t ws_size,
                              hipStream_t stream) {
  const float* X = (const float*)d_in[0];   // [256, 65536]
  const float* D = (const float*)d_in[1];   // [256, 64]
  float* Z  = (float*)d_out;                // [320, 65536]
  float* ws = (float*)d_ws;

  pm_kernel<<<1, 256, 0, stream>>>(D, ws);

  const int total = MQ * MQ + MQ * M_DIM;
  build_hw_kernel<<<(total + 255) / 256, 256, 0, stream>>>(D, ws);

  const size_t lds_bytes = (size_t)(MQ / 4) * NT * 4 * sizeof(float);  // 81920 B
  ista_kernel<<<N_DIM / NT, 640, lds_bytes, stream>>>(X, ws, Z);
}